// QuantumTSTransformer_37271726195468
// MI455X (gfx1250) — compile-verified
//
#include <hip/hip_runtime.h>
#include <math.h>

typedef float v2f __attribute__((ext_vector_type(2)));
typedef float v8f __attribute__((ext_vector_type(8)));

// D = A(16x4) * B(4x16) + C, all f32 (VOP3P 8-arg form).
#define WMMA_F32X4(a, b, c) \
    __builtin_amdgcn_wmma_f32_16x16x4_f32(false, (a), false, (b), (short)0, (c), false, false)

constexpr int BSZ = 16, TSZ = 16, FD = 64, ODIM = 8;
constexpr int NQ = 10, NAq = 4, LAYERS = 2, DEGREE = 3;
constexpr int NROTS = 80;                 // 4*NQ*LAYERS
constexpr int NSTATE = 1 << 14;           // full 14-qubit state
constexpr int NT = 1024;                  // main-kernel block size (32 waves)
constexpr float TWO_PI = 6.28318530717958647692f;

// workspace layout (floats)
constexpr int WS_TS   = 0;        // 16*16*80 = 20480
constexpr int WS_BF_R = 20480;    // 256  : B[k=a][n=j] = U[j][a]       (forward)
constexpr int WS_BF_I = 20736;    // 256
constexpr int WS_BA_R = 20992;    // 256  : B[k=a][n=j] = conj(U[a][j]) (adjoint)
constexpr int WS_BA_I = 21248;    // 256

// ---------------------------------------------------------------------------
// Kernel A: ts = sigmoid((x^T + pe) @ W_fp^T + b_fp) * 2pi     (256x64 @ 64x80)
// One wave per 16x16 output tile; K=64 in 16 wmma_f32_16x16x4 steps.
// ---------------------------------------------------------------------------
__global__ void qts_angles_kernel(const float* __restrict__ x,
                                  const float* __restrict__ Wfp,
                                  const float* __restrict__ bfp,
                                  float* __restrict__ ts)
{
    const int mt = blockIdx.x;            // 0..15 (row tiles)
    const int nt = blockIdx.y;            // 0..4  (col tiles of 80 cols)
    const int lane = threadIdx.x;         // 32 threads = one wave
    const int msel = lane & 15;
    const int kadd = (lane < 16) ? 0 : 2; // A/B lane-half K offset
    const float kdiv = -__logf(10000.0f) / (float)FD;

    const int row = mt * 16 + msel;       // row of h: row = b*16 + t
    const int bb = row >> 4;
    const int tt = row & 15;
    const int n  = nt * 16 + msel;        // output column (0..79)

    v8f acc = {};
#pragma unroll
    for (int kk = 0; kk < 16; ++kk) {
        const int k0 = kk * 4 + kadd;     // always even
        // pe[t, k0] (even->sin), pe[t, k0+1] (odd->cos); args bounded (< 16)
        const int j = k0 >> 1;
        const float d = __expf((float)(2 * j) * kdiv);
        const float arg = (float)tt * d;
        const float pe0 = __sinf(arg);
        const float pe1 = __cosf(arg);
        v2f a, b;
        a.x = x[bb * FD * TSZ + k0 * TSZ + tt] + pe0;
        a.y = x[bb * FD * TSZ + (k0 + 1) * TSZ + tt] + pe1;
        b.x = Wfp[n * FD + k0];
        b.y = Wfp[n * FD + k0 + 1];
        acc = WMMA_F32X4(a, b, acc);
    }
    const int moff = (lane < 16) ? 0 : 8;
#pragma unroll
    for (int v = 0; v < 8; ++v) {
        const int rg = mt * 16 + v + moff;           // = b*16 + t
        const int ng = nt * 16 + msel;
        const float val = acc[v] + bfp[ng];
        const float sg = 1.0f / (1.0f + __expf(-val));
        ts[rg * NROTS + ng] = sg * TWO_PI;           // ts[b][t][p]
    }
}

// ---------------------------------------------------------------------------
// Kernel B: compose the 44-gate prepare circuit into one 16x16 complex unitary
// U; write WMMA-B-layout matrices for U and U^dagger.  Thread c owns column c
// of U; gates only mix rows within a column -> no synchronization needed.
// ---------------------------------------------------------------------------
__global__ void qprep_matrix_kernel(const float* __restrict__ prep_p,
                                    float* __restrict__ ws)
{
    const int c = threadIdx.x;
    if (c >= 16) return;
    float ur[16], ui[16];
#pragma unroll
    for (int r = 0; r < 16; ++r) { ur[r] = (r == c) ? 1.f : 0.f; ui[r] = 0.f; }

    for (int ly = 0; ly < NAq; ++ly) {
        for (int qi = 0; qi < NAq; ++qi) {
            const float thy = prep_p[(ly * NAq + qi) * 2 + 0];
            const float thz = prep_p[(ly * NAq + qi) * 2 + 1];
            const int bb = 3 - qi;  // ancilla wire qi <-> bit (3-qi)
            // RY
            const float cy = __cosf(0.5f * thy), sy = __sinf(0.5f * thy);
            for (int p = 0; p < 8; ++p) {
                const int low = p & ((1 << bb) - 1);
                const int high = (p >> bb) << (bb + 1);
                const int r0 = high | low, r1 = r0 | (1 << bb);
                const float a_r = ur[r0], a_i = ui[r0], b_r = ur[r1], b_i = ui[r1];
                ur[r0] = cy * a_r - sy * b_r;  ui[r0] = cy * a_i - sy * b_i;
                ur[r1] = sy * a_r + cy * b_r;  ui[r1] = sy * a_i + cy * b_i;
            }
            // RZ: bit==0 -> *e^{-i thz/2}; bit==1 -> *e^{+i thz/2}
            const float cz = __cosf(0.5f * thz), sz = __sinf(0.5f * thz);
            for (int r = 0; r < 16; ++r) {
                const float s = ((r >> bb) & 1) ? sz : -sz;
                const float a_r = ur[r], a_i = ui[r];
                ur[r] = cz * a_r - s * a_i;
                ui[r] = cz * a_i + s * a_r;
            }
        }
        for (int i = 0; i < NAq - 1; ++i) {
            const int bc = 3 - i, bt = 3 - (i + 1);
            for (int r = 0; r < 16; ++r) {
                if (((r >> bc) & 1) && !((r >> bt) & 1)) {
                    const int r2 = r | (1 << bt);
                    const float t_r = ur[r], t_i = ui[r];
                    ur[r] = ur[r2]; ui[r] = ui[r2];
                    ur[r2] = t_r;   ui[r2] = t_i;
                }
            }
        }
    }
    float* bfr = ws + WS_BF_R;  float* bfi = ws + WS_BF_I;
    float* bar = ws + WS_BA_R;  float* bai = ws + WS_BA_I;
    for (int j2 = 0; j2 < 16; ++j2) {      // Bf[k=c][n=j] = U[j][c]
        bfr[c * 16 + j2] = ur[j2];
        bfi[c * 16 + j2] = ui[j2];
    }
    for (int a = 0; a < 16; ++a) {         // Ba[k=a][n=c] = conj(U[a][c])
        bar[a * 16 + c] = ur[a];
        bai[a * 16 + c] = -ui[a];
    }
}

// ---------------------------------------------------------------------------
// Main kernel: one workgroup (1024 thr = 32 waves) per batch element.
// Full 2^14 complex state lives in LDS as separate re/im planes (128 KB).
// ---------------------------------------------------------------------------
__device__ __forceinline__ void pcphase_lds(float* sRe, float* sIm, float phi, int tid)
{
    const float cp = __cosf(phi), sp = __sinf(phi);
    for (int i = tid; i < NSTATE; i += NT) {
        const float s = ((i & 15) == 0) ? sp : -sp;   // a==0 gets e^{+i phi}
        const float re = sRe[i], im = sIm[i];
        sRe[i] = cp * re - s * im;
        sIm[i] = cp * im + s * re;
    }
    __syncthreads();
}

// state(1024x16 complex) @ B where B is a 16x16 complex in WMMA-B layout regs.
__device__ __forceinline__ void prepare_lds(float* sRe, float* sIm,
                                            const v2f (&br)[4], const v2f (&bi)[4],
                                            int tid)
{
    const int wave = tid >> 5;            // 0..31 -> 2 tiles each (64 tiles)
    const int lane = tid & 31;
    const int msel = lane & 15;
    const int kadd = (lane < 16) ? 0 : 2;
    const int moff = (lane < 16) ? 0 : 8;
    for (int t = 0; t < 2; ++t) {
        const int rowbase = (wave * 2 + t) * 16;
        v2f ar[4], ai[4], nai[4];
#pragma unroll
        for (int kk = 0; kk < 4; ++kk) {
            const int base = (rowbase + msel) * 16 + kk * 4 + kadd;
            ar[kk].x  = sRe[base];     ar[kk].y  = sRe[base + 1];
            ai[kk].x  = sIm[base];     ai[kk].y  = sIm[base + 1];
            nai[kk].x = -ai[kk].x;     nai[kk].y = -ai[kk].y;
        }
        v8f cr = {}, ci = {};
#pragma unroll
        for (int kk = 0; kk < 4; ++kk) {
            cr = WMMA_F32X4(ar[kk],  br[kk], cr);   // +Ar*Br
            cr = WMMA_F32X4(nai[kk], bi[kk], cr);   // -Ai*Bi
            ci = WMMA_F32X4(ar[kk],  bi[kk], ci);   // +Ar*Bi
            ci = WMMA_F32X4(ai[kk],  br[kk], ci);   // +Ai*Br
        }
#pragma unroll
        for (int v = 0; v < 8; ++v) {
            const int idx = (rowbase + v + moff) * 16 + msel;
            sRe[idx] = cr[v];
            sIm[idx] = ci[v];
        }
    }
}

// sim14 circuit on the NQ wires; per-ancilla-column params with stride astride
// (astride=0 -> shared params).  adjoint <=> sign=-1 (reversed order, -theta).
// Each thread owns ancilla column a = tid&15 -> theta/cos/sin hoisted per gate.
__device__ __forceinline__ void sim14_lds(float* sRe, float* sIm,
                                          const float* params, int astride,
                                          int layers, float sign, int tid)
{
    const int a  = tid & 15;
    const int p0 = tid >> 4;              // 0..63
    const int ngates = layers * 40;
    for (int gi = 0; gi < ngates; ++gi) {
        const int g = (sign < 0.f) ? (ngates - 1 - gi) : gi;  // pidx == g
        const int r40 = g % 40;
        int kind, w0 = 0, w1 = 0;
        if (r40 < 10)      { kind = 0; w0 = r40; }
        else if (r40 < 20) { kind = 1; w0 = 19 - r40; w1 = (w0 + 1) % 10; }
        else if (r40 < 30) { kind = 0; w0 = r40 - 20; }
        else { const int j = r40 - 30; w0 = (j == 0) ? 9 : (j - 1);
               kind = 1; w1 = (w0 + 9) % 10; }

        const float th = sign * params[a * astride + g];
        const float cc = __cosf(0.5f * th), ss = __sinf(0.5f * th);

        if (kind == 0) {                       // RY(w0)
            const int bq = 9 - w0;
            for (int p = p0; p < 512; p += NT / 16) {
                const int low = p & ((1 << bq) - 1);
                const int high = (p >> bq) << (bq + 1);
                const int r0 = high | low, r1 = r0 | (1 << bq);
                const int i0 = r0 * 16 + a, i1 = r1 * 16 + a;
                const float re0 = sRe[i0], im0 = sIm[i0];
                const float re1 = sRe[i1], im1 = sIm[i1];
                sRe[i0] = cc * re0 - ss * re1;  sIm[i0] = cc * im0 - ss * im1;
                sRe[i1] = ss * re0 + cc * re1;  sIm[i1] = ss * im0 + cc * im1;
            }
        } else {                               // CRX(ctrl=w0, tgt=w1)
            const int bc = 9 - w0, bt = 9 - w1;
            const int b1 = bc < bt ? bc : bt, b2 = bc < bt ? bt : bc;
            const int midw = b2 - b1 - 1;
            for (int p = p0; p < 256; p += NT / 16) {
                const int low = p & ((1 << b1) - 1);
                const int xidx = p >> b1;
                const int mid = xidx & ((1 << midw) - 1);
                const int high = xidx >> midw;
                const int r = (high << (b2 + 1)) | (mid << (b1 + 1)) | low;
                const int r0 = r | (1 << bc);          // ctrl=1, tgt=0
                const int r1 = r0 | (1 << bt);         // ctrl=1, tgt=1
                const int i0 = r0 * 16 + a, i1 = r1 * 16 + a;
                const float re0 = sRe[i0], im0 = sIm[i0];
                const float re1 = sRe[i1], im1 = sIm[i1];
                // new0 = c*o0 - i*s*o1 ; new1 = -i*s*o0 + c*o1
                sRe[i0] = cc * re0 + ss * im1;  sIm[i0] = cc * im0 - ss * re1;
                sRe[i1] = cc * re1 + ss * im0;  sIm[i1] = cc * im1 - ss * re0;
            }
        }
        __syncthreads();
    }
}

__global__ void qsim_main_kernel(const float* __restrict__ ts,
                                 const float* __restrict__ ws,
                                 const float* __restrict__ sig_ang,
                                 const float* __restrict__ qff_p,
                                 const float* __restrict__ W_out,
                                 const float* __restrict__ b_out,
                                 float* __restrict__ out)
{
    extern __shared__ float smem[];
    float* sRe   = smem;                      // 16384
    float* sIm   = smem + NSTATE;             // 16384
    float* tsLoc = smem + 2 * NSTATE;         // 16*80 = 1280
    float* red   = tsLoc + TSZ * NROTS;       // 4
    float* exps  = red + 4;                   // 30

    const int b = blockIdx.x;
    const int tid = threadIdx.x;

    // Preload WMMA B-operand registers for U_prep and U_prep^dagger.
    v2f bf_r[4], bf_i[4], ba_r[4], ba_i[4];
    {
        const int lane = tid & 31;
        const int nsel = lane & 15;
        const int kadd = (lane < 16) ? 0 : 2;
#pragma unroll
        for (int kk = 0; kk < 4; ++kk) {
            const int kr = kk * 4 + kadd;
            bf_r[kk].x = ws[WS_BF_R + kr * 16 + nsel];
            bf_r[kk].y = ws[WS_BF_R + (kr + 1) * 16 + nsel];
            bf_i[kk].x = ws[WS_BF_I + kr * 16 + nsel];
            bf_i[kk].y = ws[WS_BF_I + (kr + 1) * 16 + nsel];
            ba_r[kk].x = ws[WS_BA_R + kr * 16 + nsel];
            ba_r[kk].y = ws[WS_BA_R + (kr + 1) * 16 + nsel];
            ba_i[kk].x = ws[WS_BA_I + kr * 16 + nsel];
            ba_i[kk].y = ws[WS_BA_I + (kr + 1) * 16 + nsel];
        }
    }

    for (int i = tid; i < NSTATE; i += NT) { sRe[i] = 0.f; sIm[i] = 0.f; }
    for (int i = tid; i < TSZ * NROTS; i += NT)
        tsLoc[i] = ts[b * TSZ * NROTS + i];
    __syncthreads();
    if (tid == 0) sRe[0] = 1.0f;
    __syncthreads();

    pcphase_lds(sRe, sIm, sig_ang[0], tid);
    for (int k = 0; k < DEGREE; ++k) {
        prepare_lds(sRe, sIm, bf_r, bf_i, tid);          // prepare
        __syncthreads();
        sim14_lds(sRe, sIm, tsLoc, NROTS, LAYERS,
                  (k & 1) ? -1.f : 1.f, tid);            // select (+/- adjoint)
        prepare_lds(sRe, sIm, ba_r, ba_i, tid);          // prepare^dagger
        __syncthreads();
        pcphase_lds(sRe, sIm, sig_ang[k + 1], tid);
    }
    sim14_lds(sRe, sIm, qff_p, 0, 1, 1.f, tid);          // final QFF layer

    // expectation values <X_i>, <Y_i>, <Z_i>
    for (int w = 0; w < NQ; ++w) {
        const int bq = 13 - w;
        float acr = 0.f, aci = 0.f, azz = 0.f;
        for (int g = tid; g < 8192; g += NT) {
            const int low = g & ((1 << bq) - 1);
            const int high = (g >> bq) << (bq + 1);
            const int i0 = high | low, i1 = i0 | (1 << bq);
            const float re0 = sRe[i0], im0 = sIm[i0];
            const float re1 = sRe[i1], im1 = sIm[i1];
            acr += re0 * re1 + im0 * im1;
            aci += re0 * im1 - im0 * re1;
            azz += re0 * re0 + im0 * im0 - re1 * re1 - im1 * im1;
        }
        if (tid < 3) red[tid] = 0.f;
        __syncthreads();
        atomicAdd(&red[0], acr);
        atomicAdd(&red[1], aci);
        atomicAdd(&red[2], azz);
        __syncthreads();
        if (tid == 0) {
            exps[w]          = 2.f * red[0];
            exps[NQ + w]     = 2.f * red[1];
            exps[2 * NQ + w] = red[2];
        }
        __syncthreads();
    }

    if (tid < ODIM) {
        float acc = b_out[tid];
        for (int j = 0; j < 3 * NQ; ++j)
            acc += exps[j] * W_out[tid * 3 * NQ + j];
        out[b * ODIM + tid] = acc;
    }
}

// ---------------------------------------------------------------------------
extern "C" void kernel_launch(void* const* d_in, const int* in_sizes, int n_in,
                              void* d_out, int out_size, void* d_ws, size_t ws_size,
                              hipStream_t stream)
{
    (void)in_sizes; (void)n_in; (void)out_size; (void)ws_size;
    const float* x       = (const float*)d_in[0];
    const float* W_fp    = (const float*)d_in[1];
    const float* b_fp    = (const float*)d_in[2];
    const float* prep_p  = (const float*)d_in[3];
    const float* sig_ang = (const float*)d_in[4];
    const float* qff_p   = (const float*)d_in[5];
    const float* W_out   = (const float*)d_in[6];
    const float* b_out   = (const float*)d_in[7];
    float* ws  = (float*)d_ws;
    float* out = (float*)d_out;

    const size_t smem =
        (size_t)(2 * NSTATE + TSZ * NROTS + 4 + 32) * sizeof(float); // ~136.5 KB
    hipFuncSetAttribute((const void*)qsim_main_kernel,
                        hipFuncAttributeMaxDynamicSharedMemorySize, (int)smem);

    dim3 gA(16, 5);  // 16 row-tiles x 5 col-tiles of the 256x80 angle GEMM
    qts_angles_kernel<<<gA, 32, 0, stream>>>(x, W_fp, b_fp, ws + WS_TS);
    qprep_matrix_kernel<<<1, 32, 0, stream>>>(prep_p, ws);
    qsim_main_kernel<<<16, NT, smem, stream>>>(ws + WS_TS, ws, sig_ang, qff_p,
                                               W_out, b_out, out);
}